// GATLayer_12584254177712
// MI455X (gfx1250) — compile-verified
//
#include <hip/hip_runtime.h>
#include <hip/hip_bf16.h>

#define IN_FEAT 128
#define N_HEADS 4
#define OUT_CH  32
#define HC      (N_HEADS * OUT_CH)   // 128

typedef float v2f __attribute__((ext_vector_type(2)));
typedef float v8f __attribute__((ext_vector_type(8)));

// ---------------------------------------------------------------------------
// Init: zero output accumulator, segmax = -inf, denom = 0
// ---------------------------------------------------------------------------
__global__ void gat_init(float* __restrict__ out, float* __restrict__ segmax,
                         float* __restrict__ denom, int n_nodes) {
    int i = blockIdx.x * blockDim.x + threadIdx.x;
    if (i < n_nodes * OUT_CH) out[i] = 0.0f;
    if (i < n_nodes * N_HEADS) {
        segmax[i] = -__builtin_inff();
        denom[i]  = 0.0f;
    }
}

// ---------------------------------------------------------------------------
// GEMM: h = x @ W  via V_WMMA_F32_16X16X4_F32 (fp32 in/out, full precision).
// Block = 256 threads = 8 waves. Block b covers rows [16b, 16b+16);
// wave w covers cols [16w, 16w+16). K-loop fully unrolled: 32 WMMAs.
//
// fp32 A (16x4) lane layout: lanes 0-15 hold M=lane, VGPR0=K0,VGPR1=K1;
//                            lanes 16-31 hold M=lane-16, VGPR0=K2,VGPR1=K3.
// fp32 B (4x16): row striped across lanes within a VGPR, K split at half-wave.
// fp32 C/D (16x16): VGPR i -> M=i (lanes 0-15) / M=i+8 (lanes 16-31), N=lane&15.
// ---------------------------------------------------------------------------
__global__ void __launch_bounds__(256) gat_gemm(const float* __restrict__ x,
                                                const float* __restrict__ W,
                                                float* __restrict__ h,
                                                int n_nodes) {
    const int lane = threadIdx.x & 31;
    const int wave = threadIdx.x >> 5;
    const int half = lane >> 4;            // 0 or 1
    const int l16  = lane & 15;
    const int col  = (wave << 4) + l16;    // 0..127
    int row = (blockIdx.x << 4) + l16;
    if (row >= n_nodes) row = n_nodes - 1; // clamp (keep EXEC all-ones for WMMA)

    const float* xr = x + (size_t)row * IN_FEAT;
    const int koff = half << 1;            // 0 or 2

    v8f c = {0.f, 0.f, 0.f, 0.f, 0.f, 0.f, 0.f, 0.f};
    #pragma unroll
    for (int kk = 0; kk < 32; ++kk) {
        const int k = (kk << 2) + koff;
        v2f a, b;
        a[0] = xr[k];
        a[1] = xr[k + 1];
        b[0] = W[(size_t)k * HC + col];
        b[1] = W[(size_t)(k + 1) * HC + col];
        c = __builtin_amdgcn_wmma_f32_16x16x4_f32(
                /*neg_a=*/false, a, /*neg_b=*/false, b,
                /*c_mod=*/(short)0, c, /*reuse_a=*/false, /*reuse_b=*/false);
    }

    const int rbase = (blockIdx.x << 4) + (half << 3);
    float* hp = h + (size_t)rbase * HC + col;
    if (rbase + 7 < n_nodes) {
        // full tile: unconditional, fully coalesced stores
        #pragma unroll
        for (int i = 0; i < 8; ++i) hp[(size_t)i * HC] = c[i];
    } else {
        #pragma unroll
        for (int i = 0; i < 8; ++i)
            if (rbase + i < n_nodes) hp[(size_t)i * HC] = c[i];
    }
}

// ---------------------------------------------------------------------------
// alpha_s[n,h] = dot(h[n,h,:], a_source[h,:]);  same for alpha_t
// ---------------------------------------------------------------------------
__global__ void gat_alpha(const float* __restrict__ h,
                          const float* __restrict__ a_src,
                          const float* __restrict__ a_tgt,
                          float* __restrict__ alpha_s,
                          float* __restrict__ alpha_t, int n_nodes) {
    int idx = blockIdx.x * blockDim.x + threadIdx.x;
    if (idx >= n_nodes * N_HEADS) return;
    const int n = idx >> 2, hd = idx & 3;
    const float* hp = h + (size_t)n * HC + hd * OUT_CH;
    const float* as = a_src + hd * OUT_CH;
    const float* at = a_tgt + hd * OUT_CH;
    float ss = 0.f, st = 0.f;
    #pragma unroll
    for (int c2 = 0; c2 < OUT_CH; ++c2) {
        const float v = hp[c2];
        ss += v * as[c2];
        st += v * at[c2];
    }
    alpha_s[idx] = ss;
    alpha_t[idx] = st;
}

__device__ __forceinline__ float edge_score(const float* alpha_s, const float* alpha_t,
                                            int s0, int t0, int hd) {
    const float s = alpha_s[s0 * N_HEADS + hd] + alpha_t[t0 * N_HEADS + hd];
    return s > 0.f ? s : 0.01f * s;   // leaky_relu, slope 0.01
}

// segmax[tgt,h] = max over in-edges (ordered int/uint trick, valid w/ -inf init)
__global__ void gat_edge_max(const int* __restrict__ src, const int* __restrict__ tgt,
                             const float* __restrict__ alpha_s,
                             const float* __restrict__ alpha_t,
                             float* __restrict__ segmax, int n_edges) {
    int idx = blockIdx.x * blockDim.x + threadIdx.x;
    if (idx >= n_edges * N_HEADS) return;
    const int e = idx >> 2, hd = idx & 3;
    const int s0 = src[e], t0 = tgt[e];
    const float s = edge_score(alpha_s, alpha_t, s0, t0, hd);
    float* addr = &segmax[t0 * N_HEADS + hd];
    if (s >= 0.f) atomicMax((int*)addr, __float_as_int(s));
    else          atomicMin((unsigned int*)addr, __float_as_uint(s));
}

// denom[tgt,h] += exp(score - segmax[tgt,h])   (scores recomputed; L2-resident)
__global__ void gat_edge_denom(const int* __restrict__ src, const int* __restrict__ tgt,
                               const float* __restrict__ alpha_s,
                               const float* __restrict__ alpha_t,
                               const float* __restrict__ segmax,
                               float* __restrict__ denom, int n_edges) {
    int idx = blockIdx.x * blockDim.x + threadIdx.x;
    if (idx >= n_edges * N_HEADS) return;
    const int e = idx >> 2, hd = idx & 3;
    const int s0 = src[e], t0 = tgt[e];
    const float s = edge_score(alpha_s, alpha_t, s0, t0, hd);
    const float ex = expf(s - segmax[t0 * N_HEADS + hd]);
    atomicAdd(&denom[t0 * N_HEADS + hd], ex);
}

// out[tgt,c] += 0.25 * sum_h att(e,h) * h[src, h*32+c]   (mean folded in)
__global__ void __launch_bounds__(256) gat_scatter(const int* __restrict__ src,
                                                   const int* __restrict__ tgt,
                                                   const float* __restrict__ h,
                                                   const float* __restrict__ alpha_s,
                                                   const float* __restrict__ alpha_t,
                                                   const float* __restrict__ segmax,
                                                   const float* __restrict__ denom,
                                                   float* __restrict__ out, int n_edges) {
    const int t = threadIdx.x;
    const int e = blockIdx.x * 8 + (t >> 5);
    if (e >= n_edges) return;
    const int c = t & 31;
    const int s0 = src[e], t0 = tgt[e];
    float acc = 0.f;
    #pragma unroll
    for (int hd = 0; hd < N_HEADS; ++hd) {
        const float s  = edge_score(alpha_s, alpha_t, s0, t0, hd);
        const float ex = expf(s - segmax[t0 * N_HEADS + hd]);
        const float att = ex / (denom[t0 * N_HEADS + hd] + 1e-16f);
        acc += h[(size_t)s0 * HC + hd * OUT_CH + c] * att;
    }
    atomicAdd(&out[(size_t)t0 * OUT_CH + c], 0.25f * acc);
}

// ---------------------------------------------------------------------------
extern "C" void kernel_launch(void* const* d_in, const int* in_sizes, int n_in,
                              void* d_out, int out_size, void* d_ws, size_t ws_size,
                              hipStream_t stream) {
    const float* x     = (const float*)d_in[0];
    const float* W     = (const float*)d_in[1];
    const float* a_src = (const float*)d_in[2];
    const float* a_tgt = (const float*)d_in[3];
    const int*   ei    = (const int*)d_in[4];

    const int n_nodes = in_sizes[0] / IN_FEAT;
    const int n_edges = in_sizes[4] / 2;
    const int* src = ei;
    const int* tgt = ei + n_edges;

    // workspace partition (h: 51.2 MB, 4 per-node arrays: 1.6 MB each)
    char* ws = (char*)d_ws;
    float* h       = (float*)ws;
    float* alpha_s = (float*)(ws + (size_t)n_nodes * HC * sizeof(float));
    float* alpha_t = alpha_s + (size_t)n_nodes * N_HEADS;
    float* segmax  = alpha_t + (size_t)n_nodes * N_HEADS;
    float* denom   = segmax  + (size_t)n_nodes * N_HEADS;
    float* out = (float*)d_out;

    // 1) init accumulators
    {
        int n = n_nodes * OUT_CH;
        gat_init<<<(n + 255) / 256, 256, 0, stream>>>(out, segmax, denom, n_nodes);
    }
    // 2) h = x @ W  (WMMA fp32)
    {
        int mblocks = (n_nodes + 15) / 16;
        gat_gemm<<<mblocks, 256, 0, stream>>>(x, W, h, n_nodes);
    }
    // 3) alpha projections
    {
        int n = n_nodes * N_HEADS;
        gat_alpha<<<(n + 255) / 256, 256, 0, stream>>>(h, a_src, a_tgt, alpha_s, alpha_t, n_nodes);
    }
    // 4) segment max
    {
        int n = n_edges * N_HEADS;
        gat_edge_max<<<(n + 255) / 256, 256, 0, stream>>>(src, tgt, alpha_s, alpha_t, segmax, n_edges);
    }
    // 5) segment sum of exp
    {
        int n = n_edges * N_HEADS;
        gat_edge_denom<<<(n + 255) / 256, 256, 0, stream>>>(src, tgt, alpha_s, alpha_t, segmax, denom, n_edges);
    }
    // 6) weighted scatter + head mean
    {
        int blocks = (n_edges + 7) / 8;
        gat_scatter<<<blocks, 256, 0, stream>>>(src, tgt, h, alpha_s, alpha_t, segmax, denom, out, n_edges);
    }
}